// FreqDBP_32091995635910
// MI455X (gfx1250) — compile-verified
//
#include <hip/hip_runtime.h>
#include <math.h>

#define LFFT   16384
#define NB     2
#define KNL    339
#define TILE_L 256
#define HALO   40
#define TROWS  (TILE_L + 2 * HALO)   /* 336 */
#define PI_D   3.14159265358979323846
#define PI_F   3.14159265358979f

typedef __attribute__((ext_vector_type(2))) float v2f;
typedef __attribute__((ext_vector_type(8))) float v8f;

/* ---- dispersion constants (beta1 == 0 because FI == FC) ---- */
constexpr double kC    = 299792458.0;
constexpr double kFc   = kC / 1.55e-6;
constexpr double c_kms = kC / 1000.0;
constexpr double lambk = c_kms / kFc;                                   /* km */
constexpr double beta2 = -(16.5 * lambk * lambk) / (2.0 * PI_D * c_kms) / 1000.0;
constexpr double dzs   = -(2000000.0 / 3.0);
constexpr double W0c   = 2.0 * PI_D * 1.6e11 / (double)LFFT;
constexpr double PC2   = -(beta2 * 0.5) * W0c * W0c * dzs;              /* phase = PC2 * fj^2 */

static __device__ __forceinline__ float2 cmulf2(float2 a, float2 b) {
    return make_float2(a.x * b.x - a.y * b.y, a.x * b.y + a.y * b.x);
}
static __device__ __forceinline__ float2 cmulcf2(float2 a, float2 b) {  /* a * conj(b) */
    return make_float2(a.x * b.x + a.y * b.y, a.y * b.x - a.x * b.y);
}

static __device__ __forceinline__ v8f wmma4(v2f a, v2f b, v8f c) {
    /* D(16x16) = A(16x4) * B(4x16) + C */
    return __builtin_amdgcn_wmma_f32_16x16x4_f32(false, a, false, b, (short)0, c, false, false);
}

/* ------------------------------------------------------------------ */
/* One in-place Stockham radix-16 stage over 16384 points in LDS.     */
/* MSH = log2(m_); TW = false when all twiddles are identity (j==0).  */
/* Fre/Fim: per-lane chunks of the constant DFT-16 matrix.            */
/* ------------------------------------------------------------------ */
template <int DIR, int MSH, bool TW>
static __device__ __forceinline__ void radix16_stage(float* sre, float* sim,
                                                     const v2f (&Fre)[4],
                                                     const v2f (&Fim)[4]) {
    const int   lane = threadIdx.x & 31;
    const int   wave = threadIdx.x >> 5;
    const int   c    = lane & 15;
    const int   h    = lane >> 4;
    const float dirf = (float)DIR;

    v8f accre[2], accim[2];
    int bfs[2];

    /* ---- read + compute (all LDS reads complete before barrier) ---- */
#pragma unroll
    for (int gi = 0; gi < 2; ++gi) {
        const int bf = (2 * wave + gi) * 16 + c;   /* butterfly id = k + m_*j */
        bfs[gi]      = bf;
        const int jm = (bf >> MSH) << MSH;          /* j * m_ */
        v8f are = {0.f, 0.f, 0.f, 0.f, 0.f, 0.f, 0.f, 0.f};
        v8f aim = {0.f, 0.f, 0.f, 0.f, 0.f, 0.f, 0.f, 0.f};
#pragma unroll
        for (int kc = 0; kc < 4; ++kc) {
            const int r0 = 4 * kc + 2 * h;          /* rows this lane holds (A and B) */
            const int r1 = r0 + 1;
            const float x0r = sre[bf + 1024 * r0], x0i = sim[bf + 1024 * r0];
            const float x1r = sre[bf + 1024 * r1], x1i = sim[bf + 1024 * r1];
            v2f Xre, Xim;
            if (TW) {
                /* input twiddle exp(i*dir*2pi*(j*m_*r)/N) */
                float s0, c0, s1, c1;
                const float tw0 = dirf * (2.0f * PI_F / (float)LFFT) * (float)((jm * r0) & (LFFT - 1));
                const float tw1 = dirf * (2.0f * PI_F / (float)LFFT) * (float)((jm * r1) & (LFFT - 1));
                __sincosf(tw0, &s0, &c0);
                __sincosf(tw1, &s1, &c1);
                Xre.x = x0r * c0 - x0i * s0;  Xim.x = x0r * s0 + x0i * c0;
                Xre.y = x1r * c1 - x1i * s1;  Xim.y = x1r * s1 + x1i * c1;
            } else {
                Xre.x = x0r;  Xim.x = x0i;
                Xre.y = x1r;  Xim.y = x1i;
            }
            const v2f Anim = {-Fim[kc].x, -Fim[kc].y};
            /* complex D = F*X : Dre = Fre*Xre - Fim*Xim ; Dim = Fim*Xre + Fre*Xim */
            are = wmma4(Fre[kc], Xre, are);
            are = wmma4(Anim,    Xim, are);
            aim = wmma4(Fim[kc], Xre, aim);
            aim = wmma4(Fre[kc], Xim, aim);
        }
        accre[gi] = are;
        accim[gi] = aim;
    }
    __syncthreads();
    /* ---- write phase: y[k + m_*(16j + u)] ---- */
#pragma unroll
    for (int gi = 0; gi < 2; ++gi) {
        const int bf = bfs[gi];
        const int j  = bf >> MSH;
        const int k  = bf & ((1 << MSH) - 1);
        const int ob = k + (j << (MSH + 4));
#pragma unroll
        for (int i = 0; i < 8; ++i) {
            const int u = i + 8 * h;                /* C/D layout: M = i + 8*half */
            sre[ob + (u << MSH)] = accre[gi][i];
            sim[ob + (u << MSH)] = accim[gi][i];
        }
    }
    __syncthreads();
}

/* ------------------------------------------------------------------ */
/* In-place 16384-pt Stockham FFT in LDS: 3 radix-16 stages (WMMA) +  */
/* one radix-4 stage. DIR = -1 forward, +1 inverse (unscaled).        */
/* ------------------------------------------------------------------ */
template <int DIR>
static __device__ void fft_core(float* sre, float* sim) {
    const int lane = threadIdx.x & 31;
    const int c    = lane & 15;
    const int h    = lane >> 4;
    const float dirf = (float)DIR;

    /* constant DFT-16 matrix chunks for this lane: F[u=c][r], 8 sincos total */
    v2f Fre[4], Fim[4];
#pragma unroll
    for (int kc = 0; kc < 4; ++kc) {
        const int r0 = 4 * kc + 2 * h;
        const int r1 = r0 + 1;
        float s0, c0, s1, c1;
        const float fa0 = dirf * (2.0f * PI_F / 16.0f) * (float)((c * r0) & 15);
        const float fa1 = dirf * (2.0f * PI_F / 16.0f) * (float)((c * r1) & 15);
        __sincosf(fa0, &s0, &c0);
        __sincosf(fa1, &s1, &c1);
        Fre[kc].x = c0;  Fre[kc].y = c1;
        Fim[kc].x = s0;  Fim[kc].y = s1;
    }

    radix16_stage<DIR, 10, false>(sre, sim, Fre, Fim);  /* m_=1024, j==0: no twiddle */
    radix16_stage<DIR, 6,  true >(sre, sim, Fre, Fim);  /* m_=64  */
    radix16_stage<DIR, 2,  true >(sre, sim, Fre, Fim);  /* m_=4   */

    /* ---- final radix-4 stage: l = 4096, m_ = 1 ---- */
    {
        const int t = threadIdx.x;
        float yr[4][4], yi[4][4];
#pragma unroll
        for (int q = 0; q < 4; ++q) {
            const int j = t + 1024 * q;
            float xr[4], xi[4];
#pragma unroll
            for (int r = 0; r < 4; ++r) {
                const float ar  = sre[j + 4096 * r];
                const float ai  = sim[j + 4096 * r];
                const float ang = dirf * (2.0f * PI_F / (float)LFFT) * (float)(j * r);
                float s, cc;
                __sincosf(ang, &s, &cc);
                xr[r] = ar * cc - ai * s;
                xi[r] = ar * s + ai * cc;
            }
            const float t0r = xr[0] + xr[2], t0i = xi[0] + xi[2];
            const float t1r = xr[0] - xr[2], t1i = xi[0] - xi[2];
            const float t2r = xr[1] + xr[3], t2i = xi[1] + xi[3];
            const float t3r = xr[1] - xr[3], t3i = xi[1] - xi[3];
            const float hr = -dirf * t3i, hi = dirf * t3r;   /* dir * i * t3 */
            yr[q][0] = t0r + t2r; yi[q][0] = t0i + t2i;
            yr[q][2] = t0r - t2r; yi[q][2] = t0i - t2i;
            yr[q][1] = t1r + hr;  yi[q][1] = t1i + hi;
            yr[q][3] = t1r - hr;  yi[q][3] = t1i - hi;
        }
        __syncthreads();
#pragma unroll
        for (int q = 0; q < 4; ++q) {
            const int j = t + 1024 * q;
#pragma unroll
            for (int u = 0; u < 4; ++u) {
                sre[4 * j + u] = yr[q][u];
                sim[4 * j + u] = yi[q][u];
            }
        }
        __syncthreads();
    }
}

/* x layout: [B][L][NMODES] interleaved complex; xf layout: [B][NMODES][L] */
__global__ __launch_bounds__(1024) void fft_fwd_kernel(const float2* __restrict__ xin,
                                                       float2* __restrict__ xf) {
    __shared__ float sre[LFFT];
    __shared__ float sim[LFFT];
    const int b = blockIdx.x >> 1;
    const int m = blockIdx.x & 1;
    for (int i = threadIdx.x; i < LFFT; i += 1024) {
        const float2 v = xin[(size_t)(b * LFFT + i) * 2 + m];
        sre[i] = v.x;
        sim[i] = v.y;
    }
    __syncthreads();
    fft_core<-1>(sre, sim);
    for (int i = threadIdx.x; i < LFFT; i += 1024)
        xf[(size_t)(b * 2 + m) * LFFT + i] = make_float2(sre[i], sim[i]);
}

__global__ __launch_bounds__(1024) void ifft_crop_kernel(const float2* __restrict__ xf,
                                                         float2* __restrict__ out) {
    __shared__ float sre[LFFT];
    __shared__ float sim[LFFT];
    const int b = blockIdx.x >> 1;
    const int m = blockIdx.x & 1;
    for (int i = threadIdx.x; i < LFFT; i += 1024) {
        const float2 v = xf[(size_t)(b * 2 + m) * LFFT + i];
        sre[i] = v.x;
        sim[i] = v.y;
    }
    __syncthreads();
    fft_core<1>(sre, sim);
    const float sc = 1.0f / (float)LFFT;
    for (int i = 50 + threadIdx.x; i < LFFT - 50; i += 1024)
        out[(size_t)(b * (LFFT - 100) + (i - 50)) * 2 + m] = make_float2(sre[i] * sc, sim[i] * sc);
}

/* (n1,n2) pairs in the reference's loop order */
__global__ void init_pairs_kernel(int* __restrict__ pairs) {
    if (threadIdx.x == 0 && blockIdx.x == 0) {
        int k = 0;
        for (int n1 = -20; n1 < 20; ++n1)
            for (int n2 = -20; n2 < 20; ++n2) {
                int p = n1 * n2;
                if (p < 0) p = -p;
                if (p <= 20) {
                    pairs[2 * k]     = n1;
                    pairs[2 * k + 1] = n2;
                    ++k;
                }
            }
    }
}

/* fused: xf_disp = xf * DK ; xout = xf_disp + (einsum + b) * P */
__global__ __launch_bounds__(256) void step_kernel(const float2* __restrict__ xin,
                                                   float2* __restrict__ xout,
                                                   const float* __restrict__ tinfo,
                                                   const float2* __restrict__ Wg,
                                                   const float2* __restrict__ bg,
                                                   const int* __restrict__ pairs) {
    __shared__ float2 tile[2][TROWS];
    __shared__ float2 Ws[KNL];
    __shared__ int    p1s[KNL];
    __shared__ int    p2s[KNL];

    const int b   = blockIdx.x >> 6;          /* 64 tiles per batch */
    const int l0  = (blockIdx.x & 63) * TILE_L;
    const int tid = threadIdx.x;

    for (int k = tid; k < KNL; k += 256) {
        p1s[k] = pairs[2 * k];
        p2s[k] = pairs[2 * k + 1];
        Ws[k]  = Wg[k];
    }
    /* halo load with dispersion applied per frequency bin */
    for (int idx = tid; idx < TROWS; idx += 256) {
        const int row = (l0 - HALO + idx) & (LFFT - 1);
        const int fj  = (row < LFFT / 2) ? row : row - LFFT;
        const double ph = PC2 * ((double)fj * (double)fj);
        const float  cp = (float)cos(ph);
        const float  sp = (float)sin(ph);
#pragma unroll
        for (int m = 0; m < 2; ++m) {
            const float2 v = xin[(size_t)(b * 2 + m) * LFFT + row];
            tile[m][idx] = make_float2(v.x * cp - v.y * sp, v.x * sp + v.y * cp);
        }
    }
    __syncthreads();

    const float  P    = 0.0005f * __powf(10.0f, tinfo[b * 4] * 0.1f);
    const float2 bb   = bg[0];
    const int    base = tid + HALO;

    float2 a0 = make_float2(0.f, 0.f);
    float2 a1 = make_float2(0.f, 0.f);
    for (int k = 0; k < KNL; ++k) {
        const int n1 = p1s[k];
        const int n2 = p2s[k];
        const float2 x10  = tile[0][base - n1];
        const float2 x11  = tile[1][base - n1];
        const float2 x120 = tile[0][base - n1 - n2];
        const float2 x121 = tile[1][base - n1 - n2];
        const float2 x20  = tile[0][base - n2];
        const float2 x21  = tile[1][base - n2];
        const float2 A0 = cmulcf2(x10, x120);
        const float2 A1 = cmulcf2(x11, x121);
        const float2 S  = make_float2(A0.x + A1.x, A0.y + A1.y);  /* A_m + A_other */
        const float2 f0 = cmulf2(S, x20);
        const float2 f1 = cmulf2(S, x21);
        const float2 w  = Ws[k];
        const float2 d0 = cmulf2(f0, w);
        const float2 d1 = cmulf2(f1, w);
        a0.x += d0.x; a0.y += d0.y;
        a1.x += d1.x; a1.y += d1.y;
    }
    const float2 xc0 = tile[0][base];
    const float2 xc1 = tile[1][base];
    const int    l   = l0 + tid;
    xout[(size_t)(b * 2 + 0) * LFFT + l] =
        make_float2(xc0.x + (a0.x + bb.x) * P, xc0.y + (a0.y + bb.y) * P);
    xout[(size_t)(b * 2 + 1) * LFFT + l] =
        make_float2(xc1.x + (a1.x + bb.x) * P, xc1.y + (a1.y + bb.y) * P);
}

extern "C" void kernel_launch(void* const* d_in, const int* in_sizes, int n_in,
                              void* d_out, int out_size, void* d_ws, size_t ws_size,
                              hipStream_t stream) {
    (void)in_sizes; (void)n_in; (void)out_size; (void)ws_size;
    const float2* x  = (const float2*)d_in[0];
    const float*  ti = (const float*)d_in[1];
    const float2* W  = (const float2*)d_in[2];
    const float2* bb = (const float2*)d_in[3];

    char* ws = (char*)d_ws;
    const size_t XFB = (size_t)NB * 2 * LFFT * sizeof(float2);  /* 512 KB */
    float2* XF0   = (float2*)ws;
    float2* XF1   = (float2*)(ws + XFB);
    int*    pairs = (int*)(ws + 2 * XFB);

    init_pairs_kernel<<<1, 32, 0, stream>>>(pairs);
    fft_fwd_kernel<<<NB * 2, 1024, 0, stream>>>(x, XF0);
    step_kernel<<<NB * 64, 256, 0, stream>>>(XF0, XF1, ti, W, bb, pairs);
    step_kernel<<<NB * 64, 256, 0, stream>>>(XF1, XF0, ti, W, bb, pairs);
    step_kernel<<<NB * 64, 256, 0, stream>>>(XF0, XF1, ti, W, bb, pairs);
    ifft_crop_kernel<<<NB * 2, 1024, 0, stream>>>(XF1, (float2*)d_out);
}